// Obj_MLPDec_70428873720070
// MI455X (gfx1250) — compile-verified
//
#include <hip/hip_runtime.h>
#include <hip/hip_bf16.h>
#include <math.h>

typedef __attribute__((ext_vector_type(2))) float v2f;
typedef __attribute__((ext_vector_type(8))) float v8f;

#define BATCH 16
#define NBOX  256
#define NCLS  151
#define HDIM  512
#define EDIM  200
#define LDS_STRIDE 153   // NMS score matrix: odd stride -> conflict-free rows
#define BSTRIDE    516   // B tile stride (dwords): %4==0 (16B-aligned rows),
                         // banks = (4n+c)%64 -> conflict-free 32-lane b64 reads

// ---------------------------------------------------------------------------
// Kernel 1: obj_dists = sem_feats(4096x512) @ W_out^T(512x151) + b_out
// V_WMMA_F32_16X16X4_F32, one 16x16 D-tile per wave.
// Block = 256 threads = 8 waves stacked in M (128 rows x 16 cols).
// B tile (16 cols x 512 K = 32KB) staged once per block in LDS, shared by all
// 8 waves. Two accumulator chains hide WMMA latency. No divergence in loop.
// ---------------------------------------------------------------------------
__global__ __launch_bounds__(256) void gemm_wmma_f32(const float* __restrict__ S,
                                                     const float* __restrict__ W,
                                                     const float* __restrict__ bias,
                                                     float* __restrict__ out) {
    __shared__ float ldsB[16 * BSTRIDE];   // ~33 KB of the WGP's 320 KB

    const int t    = threadIdx.x;
    const int wave = t >> 5;
    const int lane = t & 31;
    const int half = lane >> 4;           // 0: K pair {0,1}; 1: K pair {2,3}
    const int mr   = lane & 15;

    const int m0 = blockIdx.x * 128 + wave * 16;
    const int n0 = blockIdx.y * 16;

    // ---- cooperative stage of B tile: rows n0..n0+15 of W (clamped) ----
    // 16*512 floats, float4 per thread per step: 8 steps of 256 threads.
    {
        #pragma unroll
        for (int e = t * 4; e < 16 * HDIM; e += 256 * 4) {
            const int r = e >> 9;          // row within tile
            const int k = e & (HDIM - 1);  // k offset (multiple of 4)
            const int n = n0 + r;
            const int nc = (n < NCLS) ? n : (NCLS - 1);   // clamp, no branch
            const float4 v = *(const float4*)(W + (size_t)nc * HDIM + k);
            float* dst = &ldsB[r * BSTRIDE + k];
            dst[0] = v.x; dst[1] = v.y; dst[2] = v.z; dst[3] = v.w;
        }
    }
    __syncthreads();

    // A fragment source: row (m0+mr), columns k + 2*half + {0,1}
    const float* arow = S + (size_t)(m0 + mr) * HDIM + half * 2;
    // B fragment source (LDS): col (n0+mr) -> row mr of tile
    const float* brow = &ldsB[mr * BSTRIDE + half * 2];

    v8f acc0 = {};
    v8f acc1 = {};
    #pragma unroll 4
    for (int k = 0; k < HDIM; k += 8) {
        __builtin_prefetch(arow + k + 128, 0, 3);   // global_prefetch_b8, near
        v2f a0 = *(const v2f*)(arow + k);
        v2f b0 = *(const v2f*)(brow + k);
        v2f a1 = *(const v2f*)(arow + k + 4);
        v2f b1 = *(const v2f*)(brow + k + 4);
        // 8 args: (neg_a, A, neg_b, B, c_mod, C, reuse_a, reuse_b)
        acc0 = __builtin_amdgcn_wmma_f32_16x16x4_f32(
            false, a0, false, b0, (short)0, acc0, false, false);
        acc1 = __builtin_amdgcn_wmma_f32_16x16x4_f32(
            false, a1, false, b1, (short)0, acc1, false, false);
    }

    const int n = n0 + mr;
    if (n < NCLS) {
        const float bv = bias[n];
        // D layout: VGPR i -> row = i + 8*half, col = mr
        #pragma unroll
        for (int i = 0; i < 8; ++i) {
            const int row = m0 + half * 8 + i;
            out[(size_t)row * NCLS + n] = (acc0[i] + acc1[i]) + bv;
        }
    }
}

// ---------------------------------------------------------------------------
// Kernel 2: per-batch softmax + greedy NMS labeling.
// One block per batch (16 blocks, 256 threads). The full 256x151 score matrix
// lives in LDS (~153 KB of the WGP's 320 KB), argmax via LDS tree reduction
// with jnp.argmax tie-break (smallest flat index). IoU recomputed on the fly.
// ---------------------------------------------------------------------------
__global__ __launch_bounds__(256) void nms_greedy(const float* __restrict__ dists,
                                                  const float* __restrict__ boxes,
                                                  int* __restrict__ labels) {
    extern __shared__ float smem[];
    float* d      = smem;                           // NBOX * LDS_STRIDE floats
    float* redVal = smem + NBOX * LDS_STRIDE;       // NBOX floats
    int*   redIdx = (int*)(redVal + NBOX);          // NBOX ints

    const int t = threadIdx.x;   // row (box index) owned by this thread
    const int b = blockIdx.x;    // batch

    // --- fused softmax of row t; then col 0 := -1 ---
    {
        const float* row = dists + ((size_t)b * NBOX + t) * NCLS;
        float mx = -INFINITY;
        for (int c = 0; c < NCLS; ++c) mx = fmaxf(mx, row[c]);
        float sum = 0.0f;
        for (int c = 0; c < NCLS; ++c) {
            float e = expf(row[c] - mx);
            d[t * LDS_STRIDE + c] = e;
            sum += e;
        }
        float inv = 1.0f / sum;
        for (int c = 0; c < NCLS; ++c) d[t * LDS_STRIDE + c] *= inv;
        d[t * LDS_STRIDE + 0] = -1.0f;
    }
    __syncthreads();

    for (int iter = 0; iter < NBOX; ++iter) {
        // per-row scan; strict '>' keeps first (lowest col) on ties
        float bv = d[t * LDS_STRIDE];
        int   bi = t * NCLS;
        for (int c = 1; c < NCLS; ++c) {
            float v = d[t * LDS_STRIDE + c];
            if (v > bv) { bv = v; bi = t * NCLS + c; }
        }
        redVal[t] = bv;
        redIdx[t] = bi;
        __syncthreads();

        // tree argmax across 256 rows; tie -> smaller flat index
        for (int s = 128; s > 0; s >>= 1) {
            if (t < s) {
                float v2 = redVal[t + s];
                int   i2 = redIdx[t + s];
                if (v2 > redVal[t] || (v2 == redVal[t] && i2 < redIdx[t])) {
                    redVal[t] = v2;
                    redIdx[t] = i2;
                }
            }
            __syncthreads();
        }

        const int flat = redIdx[0];
        const int box  = flat / NCLS;
        const int cls  = flat - box * NCLS;
        if (t == 0) labels[b * NBOX + box] = cls;

        // IoU(boxes[b,box,cls], boxes[b,t,cls])
        const float* pb = boxes + (((size_t)b * NBOX + box) * NCLS + cls) * 4;
        const float* pa = boxes + (((size_t)b * NBOX + t)   * NCLS + cls) * 4;
        const float bx1 = pb[0], by1 = pb[1], bx2 = pb[2], by2 = pb[3];
        const float ax1 = pa[0], ay1 = pa[1], ax2 = pa[2], ay2 = pa[3];
        const float w = fmaxf(fminf(ax2, bx2) - fmaxf(ax1, bx1) + 1.0f, 0.0f);
        const float h = fmaxf(fminf(ay2, by2) - fmaxf(ay1, by1) + 1.0f, 0.0f);
        const float inter  = w * h;
        const float areaA  = (ax2 - ax1 + 1.0f) * (ay2 - ay1 + 1.0f);
        const float areaB  = (bx2 - bx1 + 1.0f) * (by2 - by1 + 1.0f);
        const float iou    = inter / (areaA + areaB - inter);
        const bool  ov     = (iou >= 0.5f);

        __syncthreads();                          // reads of d / redIdx done
        if (ov) d[t * LDS_STRIDE + cls] = 0.0f;   // zero overlapping entries
        __syncthreads();
        if (t < NCLS) d[box * LDS_STRIDE + t] = -1.0f;  // retire chosen row
        __syncthreads();
    }
}

// ---------------------------------------------------------------------------
// Kernel 3: preds (as float) + embedding gather obj_embed[preds]
// ---------------------------------------------------------------------------
__global__ __launch_bounds__(256) void gather_embed(const float* __restrict__ obj_embed,
                                                    const int* __restrict__ labels,
                                                    float* __restrict__ out_preds,
                                                    float* __restrict__ out_embed) {
    const int tid = blockIdx.x * blockDim.x + threadIdx.x;
    if (tid < BATCH * NBOX) {
        out_preds[tid] = (float)labels[tid];
    }
    if (tid < BATCH * NBOX * EDIM) {
        const int row = tid / EDIM;
        const int e   = tid - row * EDIM;
        out_embed[tid] = obj_embed[(size_t)labels[row] * EDIM + e];
    }
}

extern "C" void kernel_launch(void* const* d_in, const int* in_sizes, int n_in,
                              void* d_out, int out_size, void* d_ws, size_t ws_size,
                              hipStream_t stream) {
    const float* sem_feats  = (const float*)d_in[0];   // (4096, 512)
    const float* boxes      = (const float*)d_in[1];   // (16, 256, 151, 4)
    const float* W_out      = (const float*)d_in[2];   // (151, 512)
    const float* b_out      = (const float*)d_in[3];   // (151,)
    const float* obj_embed  = (const float*)d_in[4];   // (151, 200)

    float* out = (float*)d_out;
    float* out_dists = out;                                  // 4096*151
    float* out_preds = out + (size_t)BATCH * NBOX * NCLS;    // 4096
    float* out_embed = out_preds + (size_t)BATCH * NBOX;     // 4096*200

    int* labels = (int*)d_ws;                                // 4096 ints scratch

    // 1) WMMA GEMM: obj_dists
    dim3 ggrid(32, (NCLS + 15) / 16);
    gemm_wmma_f32<<<ggrid, 256, 0, stream>>>(sem_feats, W_out, b_out, out_dists);

    // 2) softmax + greedy NMS (one block per batch, score matrix in LDS)
    const size_t smem_bytes =
        (size_t)(NBOX * LDS_STRIDE) * sizeof(float)   // d
        + NBOX * sizeof(float)                        // redVal
        + NBOX * sizeof(int);                         // redIdx
    nms_greedy<<<BATCH, NBOX, smem_bytes, stream>>>(out_dists, boxes, labels);

    // 3) preds + embedding gather
    const int total = BATCH * NBOX * EDIM;
    gather_embed<<<(total + 255) / 256, 256, 0, stream>>>(obj_embed, labels,
                                                          out_preds, out_embed);
}